// SelfAttention_8701603741783
// MI455X (gfx1250) — compile-verified
//
#include <hip/hip_runtime.h>

typedef float v2f __attribute__((ext_vector_type(2)));
typedef float v8f __attribute__((ext_vector_type(8)));
typedef unsigned int v4u __attribute__((ext_vector_type(4)));
typedef int v4i __attribute__((ext_vector_type(4)));
typedef int v8i __attribute__((ext_vector_type(8)));

#define S_DIM 2048
#define B_DIM 16
#define F_DIM 256
#define QS 260   // Q tile LDS stride: 260%64==4 -> rows hit distinct banks
#define KS 260   // K tile LDS stride (TDM pads each 256-dword row by 4 dwords)
#define VS 264   // V tile LDS stride: 264%64==8 -> lane halves hit disjoint banks
#define SS 18    // score tile stride (even -> 8B-aligned b64 reads)
#define NW 4     // waves per block (4 query blocks share staged K/V)

__device__ __forceinline__ v8f v8f_zero() {
    v8f z;
#pragma unroll
    for (int i = 0; i < 8; ++i) z[i] = 0.0f;
    return z;
}

__device__ __forceinline__ v8f wmma_f32(v2f a, v2f b, v8f c) {
    // D = A(16x4,f32) x B(4x16,f32) + C(16x16,f32)
    return __builtin_amdgcn_wmma_f32_16x16x4_f32(
        false, a, false, b, (short)0, c, false, false);
}

// Async copy 16B/lane (512B per instruction) global -> LDS, ASYNCcnt-tracked.
__device__ __forceinline__ void async_b128(unsigned lds_off, const float* gaddr) {
    asm volatile("global_load_async_to_lds_b128 %0, %1, off"
                 :: "v"(lds_off), "v"(gaddr)
                 : "memory");
}
__device__ __forceinline__ void wait_async0() {
    asm volatile("s_wait_asynccnt 0" ::: "memory");
}
__device__ __forceinline__ unsigned lds_off(const void* p) {
    // generic LDS pointer = {shared_aperture_hi, 32-bit LDS offset}
    return (unsigned)(uintptr_t)p;
}

// TDM: move one 16x256 f32 tile (row stride 256 floats) global->LDS, padding
// each 256-dword row by 4 dwords so the LDS row stride is KS=260 floats.
__device__ __forceinline__ void tdm_load_k_tile(const float* gsrc, unsigned ldsaddr) {
    const unsigned long long ga = (unsigned long long)(uintptr_t)gsrc;
    v4u g0;
    g0[0] = 1u;                                   // count=1 (valid user D#)
    g0[1] = ldsaddr;                              // lds_addr (bytes)
    g0[2] = (unsigned)(ga & 0xffffffffu);         // global_addr[31:0]
    g0[3] = (unsigned)((ga >> 32) & 0x1ffffffu)   // global_addr[56:32]
          | (2u << 30);                           // type=2 ("image")
    v8i g1;
    g1[0] = (int)((2u << 16)                      // data_size = 4B
                | (1u << 20)                      // pad_enable
                | (7u << 22)                      // pad_interval: 256 dwords
                | (3u << 25));                    // pad_amount: 4 dwords
    g1[1] = (int)(256u << 16);                    // tensor_dim0[15:0]=256 @bits[79:64... (31:16 of w1)]
    g1[2] = (int)(16u << 16);                     // tensor_dim0 hi=0 | tensor_dim1[15:0]=16
    g1[3] = (int)(256u << 16);                    // tensor_dim1 hi=0 | tile_dim0=256
    g1[4] = 16;                                   // tile_dim1=16, tile_dim2=0
    g1[5] = 256;                                  // tensor_dim0_stride[31:0]=256
    g1[6] = 0;                                    // stride hi / tensor_dim1_stride lo
    g1[7] = 0;
    const v4i z4 = {0, 0, 0, 0};
#if __clang_major__ >= 23
    const v8i z8 = {0, 0, 0, 0, 0, 0, 0, 0};
    __builtin_amdgcn_tensor_load_to_lds(g0, g1, z4, z4, z8, 0);
#else
    __builtin_amdgcn_tensor_load_to_lds(g0, g1, z4, z4, 0);
#endif
}

// ---------------------------------------------------------------------------
// Kernel 1: QKV projection.  X = x viewed as [S*B, F] (row = s*B + b).
// B == 16, so an aligned 16-row tile is (seq s) x (all 16 batches).
// Output layout: [B][S][F].  grid=(S,3), block=128 (4 waves x 16x64 strip).
// ---------------------------------------------------------------------------
__global__ __launch_bounds__(128) void qkv_proj_kernel(
    const float* __restrict__ x,
    const float* __restrict__ Wq, const float* __restrict__ bq,
    const float* __restrict__ Wk, const float* __restrict__ bk,
    const float* __restrict__ Wv, const float* __restrict__ bv,
    float* __restrict__ Q, float* __restrict__ K, float* __restrict__ V)
{
    const int lane = threadIdx.x & 31;
    const int wave = threadIdx.x >> 5;
    const int s    = blockIdx.x;
    const int which = blockIdx.y;

    const float* W;
    const float* bias;
    float* out;
    if (which == 0)      { W = Wq; bias = bq; out = Q; }
    else if (which == 1) { W = Wk; bias = bk; out = K; }
    else                 { W = Wv; bias = bv; out = V; }

    const int n  = lane & 15;
    const int h  = lane >> 4;
    const int n0 = wave * 64;

    v8f acc[4];
#pragma unroll
    for (int t = 0; t < 4; ++t) acc[t] = v8f_zero();

    const float* aptr = x + ((size_t)s * 16 + n) * F_DIM + 2 * h;

    for (int k0 = 0; k0 < F_DIM; k0 += 4) {
        v2f a = *(const v2f*)(aptr + k0);
#pragma unroll
        for (int t = 0; t < 4; ++t) {
            const float* bp = W + (size_t)(n0 + t * 16 + n) * F_DIM + k0 + 2 * h;
            acc[t] = wmma_f32(a, *(const v2f*)bp, acc[t]);
        }
    }

#pragma unroll
    for (int t = 0; t < 4; ++t) {
        const int col = n0 + t * 16 + n;
        const float bb = bias[col];
#pragma unroll
        for (int i = 0; i < 8; ++i) {
            const int mr = i + 8 * h;  // batch index
            out[((size_t)mr * S_DIM + s) * F_DIM + col] = acc[t][i] + bb;
        }
    }
}

// ---------------------------------------------------------------------------
// Kernel 2: flash attention. grid = (S/16/NW, B), block = 32*NW.
// 4 waves (4 query blocks, same batch) share double-buffered K/V tiles:
// K staged by TDM (tensor_load_to_lds, TENSORcnt), V by async b128 (ASYNCcnt).
// ---------------------------------------------------------------------------
__global__ __launch_bounds__(32 * NW) void attn_kernel(
    const float* __restrict__ Q, const float* __restrict__ K,
    const float* __restrict__ V, float* __restrict__ out)
{
    __shared__ float q_s[NW][16 * QS];
    __shared__ float k_s[2][16 * KS];
    __shared__ float v_s[2][16 * VS];
    __shared__ float sc[NW][16 * SS];
    __shared__ float mrow[NW][16];
    __shared__ float lrow[NW][16];
    __shared__ float arow[NW][16];

    const int lane = threadIdx.x & 31;
    const int wave = threadIdx.x >> 5;
    const int b    = blockIdx.y;
    const int s0   = (blockIdx.x * NW + wave) * 16;
    const float scale = 0.0625f;  // 1/sqrt(256)

    const float* qptr = Q + ((size_t)b * S_DIM + s0) * F_DIM;
    const float* kptr = K + (size_t)b * S_DIM * F_DIM;
    const float* vptr = V + (size_t)b * S_DIM * F_DIM;

    // stage this wave's Q tile into LDS
    for (int idx = lane; idx < 16 * (F_DIM / 4); idx += 32) {
        const int r  = idx >> 6;
        const int c4 = (idx & 63) << 2;
        float4 v = *(const float4*)(qptr + (size_t)r * F_DIM + c4);
        *(float4*)&q_s[wave][r * QS + c4] = v;
    }
    if (lane < 16) { mrow[wave][lane] = -__builtin_huge_valf(); lrow[wave][lane] = 0.0f; }

    // each wave issues its quarter (8 b128 instrs) of the V tile async copy
    auto issue_v = [&](const float* vt, int buf) {
#pragma unroll
        for (int j = 0; j < 8; ++j) {
            const int idx = (wave * 8 + j) * 32 + lane;  // float4 index in 16x256
            const int row = idx >> 6;
            const int c4  = (idx & 63) << 2;
            async_b128(lds_off(&v_s[buf][row * VS + c4]), vt + (size_t)row * F_DIM + c4);
        }
    };

    if (wave == 0) tdm_load_k_tile(kptr, lds_off(&k_s[0][0]));
    issue_v(vptr, 0);
    wait_async0();
    if (wave == 0) __builtin_amdgcn_s_wait_tensorcnt(0);
    __syncthreads();

    const int n = lane & 15;
    const int h = lane >> 4;

    v8f acc[16];
#pragma unroll
    for (int t = 0; t < 16; ++t) acc[t] = v8f_zero();

    const float* qa = &q_s[wave][n * QS + 2 * h];

    int cur = 0;
    for (int t0 = 0; t0 < S_DIM; t0 += 16) {
        // prefetch next K/V tile into the other buffer (overlaps compute)
        if (t0 + 16 < S_DIM) {
            if (wave == 0)
                tdm_load_k_tile(kptr + (size_t)(t0 + 16) * F_DIM,
                                lds_off(&k_s[cur ^ 1][0]));
            issue_v(vptr + (size_t)(t0 + 16) * F_DIM, cur ^ 1);
        }

        // ---- scores: S = Q * K^T (16x16) over F=256, K tile from LDS ----
        v8f sacc = v8f_zero();
        const float* kb = &k_s[cur][n * KS + 2 * h];
        for (int k0 = 0; k0 < F_DIM; k0 += 4) {
            v2f a  = *(const v2f*)(qa + k0);
            v2f bf = *(const v2f*)(kb + k0);
            sacc = wmma_f32(a, bf, sacc);
        }
#pragma unroll
        for (int i = 0; i < 8; ++i)
            sc[wave][(i + 8 * h) * SS + n] = sacc[i] * scale;
        // wave-local LDS ops are in-order: no barrier needed before softmax

        // ---- per-row online softmax (lanes 0..15, one row each) ----
        if (lane < 16) {
            const float mold = mrow[wave][lane];
            const float lold = lrow[wave][lane];
            float mx = mold;
            float sv[16];
#pragma unroll
            for (int j = 0; j < 16; ++j) {
                sv[j] = sc[wave][lane * SS + j];
                mx = fmaxf(mx, sv[j]);
            }
            const float alpha = __expf(mold - mx);
            float ls = 0.0f;
#pragma unroll
            for (int j = 0; j < 16; ++j) {
                const float p = __expf(sv[j] - mx);
                sc[wave][lane * SS + j] = p;
                ls += p;
            }
            mrow[wave][lane] = mx;
            lrow[wave][lane] = lold * alpha + ls;
            arow[wave][lane] = alpha;
        }

        // ---- rescale O accumulators by alpha(row) ----
        float al[8];
#pragma unroll
        for (int i = 0; i < 8; ++i) al[i] = arow[wave][i + 8 * h];
#pragma unroll
        for (int t = 0; t < 16; ++t) {
#pragma unroll
            for (int i = 0; i < 8; ++i) acc[t][i] *= al[i];
        }

        // ---- O += P(16x16) * V(16x256), V tile from LDS ----
        const float* vcur = v_s[cur];
#pragma unroll
        for (int kk = 0; kk < 4; ++kk) {
            v2f a = *(const v2f*)&sc[wave][n * SS + kk * 4 + 2 * h];
            const int r0 = kk * 4 + 2 * h;
#pragma unroll
            for (int t = 0; t < 16; ++t) {
                v2f bf;
                bf.x = vcur[r0 * VS + t * 16 + n];
                bf.y = vcur[(r0 + 1) * VS + t * 16 + n];
                acc[t] = wmma_f32(a, bf, acc[t]);
            }
        }

        // prefetches landed + all waves done reading `cur`
        wait_async0();
        if (wave == 0) __builtin_amdgcn_s_wait_tensorcnt(0);
        __syncthreads();
        cur ^= 1;
    }

    // ---- normalize by l and store to [S][B][F] ----
    float li[8];
#pragma unroll
    for (int i = 0; i < 8; ++i) li[i] = 1.0f / lrow[wave][i + 8 * h];
#pragma unroll
    for (int t = 0; t < 16; ++t) {
        const int col = t * 16 + n;
#pragma unroll
        for (int i = 0; i < 8; ++i) {
            const int mr = i + 8 * h;
            out[(((size_t)(s0 + mr)) * B_DIM + b) * F_DIM + col] = acc[t][i] * li[i];
        }
    }
}

// ---------------------------------------------------------------------------
extern "C" void kernel_launch(void* const* d_in, const int* in_sizes, int n_in,
                              void* d_out, int out_size, void* d_ws, size_t ws_size,
                              hipStream_t stream) {
    const float* x  = (const float*)d_in[0];
    const float* Wq = (const float*)d_in[1];
    const float* bq = (const float*)d_in[2];
    const float* Wk = (const float*)d_in[3];
    const float* bk = (const float*)d_in[4];
    const float* Wv = (const float*)d_in[5];
    const float* bv = (const float*)d_in[6];
    float* out = (float*)d_out;

    const size_t SBF = (size_t)S_DIM * B_DIM * F_DIM;
    float* Q = (float*)d_ws;
    float* K = Q + SBF;
    float* V = K + SBF;

    dim3 g1(S_DIM, 3);
    qkv_proj_kernel<<<g1, 128, 0, stream>>>(x, Wq, bq, Wk, bk, Wv, bv, Q, K, V);

    dim3 g2(S_DIM / 16 / NW, B_DIM);
    attn_kernel<<<g2, 32 * NW, 0, stream>>>(Q, K, V, out);
}